// forward_warp_rescalled_3092376453252
// MI455X (gfx1250) — compile-verified
//
#include <hip/hip_runtime.h>

// Forward-warp bilinear splat + normalize for MI455X (gfx1250).
//
// No matmul -> no WMMA. Optimization is memory-system driven:
//  - accumulator (132.7 MB) fits in the 192 MB L2; atomics at agent scope
//    resolve at L2 (global_atomic_add_f32, non-returning form).
//  - streaming inputs loaded non-temporal so they don't evict the
//    L2-resident accumulator; output stored non-temporal.
//  - AoS float4 accumulator {c0,c1,c2,w}: each corner's 4 atomics hit one
//    16B line instead of 3 planes 2MB apart.

typedef float v2f __attribute__((ext_vector_type(2)));
typedef float v4f __attribute__((ext_vector_type(4)));

static constexpr int Bn = 16, Cn = 3, Hn = 540, Wn = 960;
static constexpr int HW = Hn * Wn;           // 518400
static constexpr int Npix = Bn * HW;         // 8,294,400  (fits in int32)
static constexpr float EPSf = 1e-6f;

__device__ __forceinline__ void fadd_agent(float* p, float v) {
    (void)__hip_atomic_fetch_add(p, v, __ATOMIC_RELAXED, __HIP_MEMORY_SCOPE_AGENT);
}

// ---------------------------------------------------------------- zeroing
__global__ void zero_f4_kernel(v4f* __restrict__ p, int n4) {
    const int stride = gridDim.x * blockDim.x;
    v4f z = {0.f, 0.f, 0.f, 0.f};
    for (int t = blockIdx.x * blockDim.x + threadIdx.x; t < n4; t += stride)
        p[t] = z;
}

// ---------------------------------------------------------------- AoS path
// acc layout: acc[b*HW + y*W + x] = {c0, c1, c2, weight}
__global__ void splat_aos_kernel(const float* __restrict__ im0,
                                 const float* __restrict__ flow,
                                 float* __restrict__ acc) {
    const int stride = gridDim.x * blockDim.x;
    for (int t = blockIdx.x * blockDim.x + threadIdx.x; t < Npix; t += stride) {
        int tn = t + stride;
        if (tn < Npix) {
            // gfx1250 global_prefetch_b8 for the next iteration's flow
            __builtin_prefetch(flow + 2 * (size_t)tn, 0, 0);
        }
        const int b = t / HW;
        const int p = t - b * HW;
        const int h = p / Wn;
        const int w = p - h * Wn;

        // streaming, read-once -> non-temporal (protect L2 accumulator)
        const float fx = __builtin_nontemporal_load(flow + 2 * (size_t)t);
        const float fy = __builtin_nontemporal_load(flow + 2 * (size_t)t + 1);
        const float* src = im0 + (size_t)b * Cn * HW + p;
        const float v0 = __builtin_nontemporal_load(src);
        const float v1 = __builtin_nontemporal_load(src + HW);
        const float v2 = __builtin_nontemporal_load(src + 2 * HW);

        const float x = (float)w + fx;
        const float y = (float)h + fy;
        const float x0 = floorf(x);
        const float y0 = floorf(y);
        const float wx1 = x - x0, wx0 = 1.0f - wx1;
        const float wy1 = y - y0, wy0 = 1.0f - wy1;

        float* accb = acc + (size_t)b * HW * 4;
        const float Wm1 = (float)(Wn - 1), Hm1 = (float)(Hn - 1);

        auto corner = [&](float cx, float cy, float wt) {
            if (cx >= 0.f && cx <= Wm1 && cy >= 0.f && cy <= Hm1) {
                const int ix = (int)cx;
                const int iy = (int)cy;
                float* q = accb + 4 * (size_t)(iy * Wn + ix);
                fadd_agent(q + 0, v0 * wt);
                fadd_agent(q + 1, v1 * wt);
                fadd_agent(q + 2, v2 * wt);
                fadd_agent(q + 3, wt);
            }
        };
        corner(x0,        y0,        wx0 * wy0);
        corner(x0 + 1.0f, y0,        wx1 * wy0);
        corner(x0,        y0 + 1.0f, wx0 * wy1);
        corner(x0 + 1.0f, y0 + 1.0f, wx1 * wy1);
    }
}

__global__ void norm_aos_kernel(const v4f* __restrict__ acc,
                                float* __restrict__ out) {
    const int stride = gridDim.x * blockDim.x;
    for (int t = blockIdx.x * blockDim.x + threadIdx.x; t < Npix; t += stride) {
        const v4f a = acc[t];                       // global_load_b128
        const float m = (a.w < EPSf) ? 1.0f : a.w;
        const float inv = 1.0f / m;
        const int b = t / HW;
        const int p = t - b * HW;
        float* o = out + (size_t)b * Cn * HW + p;
        __builtin_nontemporal_store(a.x * inv, o);
        __builtin_nontemporal_store(a.y * inv, o + HW);
        __builtin_nontemporal_store(a.z * inv, o + 2 * HW);
    }
}

// ------------------------------------------------------------- planar path
// (fallback if ws too small for AoS): warped accumulates in d_out (planar),
// mask weights accumulate in ws[Npix].
__global__ void splat_planar_kernel(const float* __restrict__ im0,
                                    const float* __restrict__ flow,
                                    float* __restrict__ out,
                                    float* __restrict__ maskacc) {
    const int stride = gridDim.x * blockDim.x;
    for (int t = blockIdx.x * blockDim.x + threadIdx.x; t < Npix; t += stride) {
        int tn = t + stride;
        if (tn < Npix) __builtin_prefetch(flow + 2 * (size_t)tn, 0, 0);

        const int b = t / HW;
        const int p = t - b * HW;
        const int h = p / Wn;
        const int w = p - h * Wn;

        const float fx = __builtin_nontemporal_load(flow + 2 * (size_t)t);
        const float fy = __builtin_nontemporal_load(flow + 2 * (size_t)t + 1);
        const float* src = im0 + (size_t)b * Cn * HW + p;
        const float v0 = __builtin_nontemporal_load(src);
        const float v1 = __builtin_nontemporal_load(src + HW);
        const float v2 = __builtin_nontemporal_load(src + 2 * HW);

        const float x = (float)w + fx;
        const float y = (float)h + fy;
        const float x0 = floorf(x);
        const float y0 = floorf(y);
        const float wx1 = x - x0, wx0 = 1.0f - wx1;
        const float wy1 = y - y0, wy0 = 1.0f - wy1;

        float* outb = out + (size_t)b * Cn * HW;
        float* mskb = maskacc + (size_t)b * HW;
        const float Wm1 = (float)(Wn - 1), Hm1 = (float)(Hn - 1);

        auto corner = [&](float cx, float cy, float wt) {
            if (cx >= 0.f && cx <= Wm1 && cy >= 0.f && cy <= Hm1) {
                const int idx = (int)cy * Wn + (int)cx;
                fadd_agent(outb + idx,          v0 * wt);
                fadd_agent(outb + HW + idx,     v1 * wt);
                fadd_agent(outb + 2 * HW + idx, v2 * wt);
                fadd_agent(mskb + idx,          wt);
            }
        };
        corner(x0,        y0,        wx0 * wy0);
        corner(x0 + 1.0f, y0,        wx1 * wy0);
        corner(x0,        y0 + 1.0f, wx0 * wy1);
        corner(x0 + 1.0f, y0 + 1.0f, wx1 * wy1);
    }
}

__global__ void norm_planar_kernel(const float* __restrict__ maskacc,
                                   float* __restrict__ out) {
    const int stride = gridDim.x * blockDim.x;
    for (int t = blockIdx.x * blockDim.x + threadIdx.x; t < Npix; t += stride) {
        const float m0 = maskacc[t];
        const float inv = 1.0f / ((m0 < EPSf) ? 1.0f : m0);
        const int b = t / HW;
        const int p = t - b * HW;
        float* o = out + (size_t)b * Cn * HW + p;
        o[0]      *= inv;
        o[HW]     *= inv;
        o[2 * HW] *= inv;
    }
}

// ---------------------------------------------------------------- launch
extern "C" void kernel_launch(void* const* d_in, const int* in_sizes, int n_in,
                              void* d_out, int out_size, void* d_ws, size_t ws_size,
                              hipStream_t stream) {
    const float* im0  = (const float*)d_in[0];
    const float* flow = (const float*)d_in[1];
    float* out = (float*)d_out;

    const int blocks = 8192, threads = 256;   // wave32: 8 waves/block
    const size_t aos_bytes = (size_t)Npix * 16;

    if (ws_size >= aos_bytes) {
        float* acc = (float*)d_ws;
        zero_f4_kernel<<<blocks, threads, 0, stream>>>((v4f*)acc, Npix);
        splat_aos_kernel<<<blocks, threads, 0, stream>>>(im0, flow, acc);
        norm_aos_kernel<<<blocks, threads, 0, stream>>>((const v4f*)acc, out);
    } else {
        float* maskacc = (float*)d_ws;
        // zero output (3*Npix floats) and mask (Npix floats); both /4 exact
        zero_f4_kernel<<<blocks, threads, 0, stream>>>((v4f*)out, (3 * Npix) / 4);
        zero_f4_kernel<<<blocks, threads, 0, stream>>>((v4f*)maskacc, Npix / 4);
        splat_planar_kernel<<<blocks, threads, 0, stream>>>(im0, flow, out, maskacc);
        norm_planar_kernel<<<blocks, threads, 0, stream>>>(maskacc, out);
    }
}